// TunedModel_56014963474648
// MI455X (gfx1250) — compile-verified
//
#include <hip/hip_runtime.h>
#include <hip/hip_bf16.h>
#include <math.h>

typedef __attribute__((ext_vector_type(2))) float v2f;
typedef __attribute__((ext_vector_type(8))) float v8f;

__device__ inline v8f zero8() {
  v8f v;
#pragma unroll
  for (int i = 0; i < 8; ++i) v[i] = 0.f;
  return v;
}

__device__ inline v8f wmma4(v2f a, v2f b, v8f c) {
  // D = A(16x4,f32) x B(4x16,f32) + C(16x16,f32)
  return __builtin_amdgcn_wmma_f32_16x16x4_f32(false, a, false, b, (short)0, c,
                                               false, false);
}

__device__ inline v2f ld2(const float* p) { return *(const v2f*)p; }

__device__ inline void atomicMaxF(float* addr, float val) {
  unsigned int* ua = (unsigned int*)addr;
  unsigned int old = *ua;
  while (__uint_as_float(old) < val) {
    unsigned int assumed = old;
    old = atomicCAS(ua, assumed, __float_as_uint(val));
    if (old == assumed) break;
  }
}

__device__ inline float sigmoidf(float x) { return 1.f / (1.f + expf(-x)); }
__device__ inline float geluf(float x) {
  return 0.5f * x * (1.f + erff(x * 0.70710678118654752f));
}

// ---------------------------------------------------------------------------
// fill
// ---------------------------------------------------------------------------
__global__ void fill_kernel(float* __restrict__ p, long long cnt, float v) {
  long long i = (long long)blockIdx.x * blockDim.x + threadIdx.x;
  if (i < cnt) p[i] = v;
}

// ---------------------------------------------------------------------------
// C[M,N] = A[M,K] @ B[K,N] (+bias) (+= old C if acc).
// REQUIRES: M % 32 == 0, N % 64 == 0, K % 4 == 0 (true for all call sites).
// One wave computes a 32x64 block of C (8 f32 WMMA accumulators).
// Software-pipelined: the next K-step's A/B operands are prefetched into
// registers before the current step's 8 WMMAs issue, so VMEM overlaps the
// matrix pipe instead of s_wait_loadcnt 0x0 before every WMMA pair.
// ---------------------------------------------------------------------------
__global__ __launch_bounds__(128) void gemm_wmma_kernel(
    const float* __restrict__ A, int lda, const float* __restrict__ B, int ldb,
    float* __restrict__ C, int ldc, int M, int N, int K,
    const float* __restrict__ bias, int accflag) {
  const int wave = threadIdx.x >> 5;
  const int lane = threadIdx.x & 31;
  const int m0 = (blockIdx.x * 4 + wave) * 32;
  const int n0 = blockIdx.y * 64;
  if (m0 >= M) return;
  const int hi = lane >> 4;  // half-wave selects K offset 0/2
  const int l15 = lane & 15;
  const float* ap0 = A + (size_t)(m0 + l15) * lda + hi * 2;
  const float* ap1 = A + (size_t)(m0 + 16 + l15) * lda + hi * 2;
  const float* bp = B + (size_t)(hi * 2) * ldb + n0 + l15;

  v8f acc[8];
#pragma unroll
  for (int i = 0; i < 8; ++i) acc[i] = zero8();

  // pipeline registers: current K-step operands
  v2f a0c = ld2(ap0);
  v2f a1c = ld2(ap1);
  v2f bc[4];
#pragma unroll
  for (int tn = 0; tn < 4; ++tn) {
    bc[tn].x = bp[tn * 16];
    bc[tn].y = bp[(size_t)ldb + tn * 16];
  }

  for (int k = 0; k < K - 4; k += 4) {
    // prefetch next K-step
    const v2f a0n = ld2(ap0 + k + 4);
    const v2f a1n = ld2(ap1 + k + 4);
    const float* bpn = bp + (size_t)(k + 4) * ldb;
    v2f bn[4];
#pragma unroll
    for (int tn = 0; tn < 4; ++tn) {
      bn[tn].x = bpn[tn * 16];
      bn[tn].y = bpn[(size_t)ldb + tn * 16];
    }
    // compute on current operands (overlaps the prefetch loads)
#pragma unroll
    for (int tn = 0; tn < 4; ++tn) {
      acc[tn] = wmma4(a0c, bc[tn], acc[tn]);
      acc[4 + tn] = wmma4(a1c, bc[tn], acc[4 + tn]);
    }
    a0c = a0n;
    a1c = a1n;
#pragma unroll
    for (int tn = 0; tn < 4; ++tn) bc[tn] = bn[tn];
  }
  // epilogue: last K-step
#pragma unroll
  for (int tn = 0; tn < 4; ++tn) {
    acc[tn] = wmma4(a0c, bc[tn], acc[tn]);
    acc[4 + tn] = wmma4(a1c, bc[tn], acc[4 + tn]);
  }

  float bb[4];
#pragma unroll
  for (int tn = 0; tn < 4; ++tn) bb[tn] = bias ? bias[n0 + tn * 16 + l15] : 0.f;

#pragma unroll
  for (int half = 0; half < 2; ++half) {
#pragma unroll
    for (int r = 0; r < 8; ++r) {
      const int row = m0 + half * 16 + r + hi * 8;
      float* cp = C + (size_t)row * ldc + n0 + l15;
#pragma unroll
      for (int tn = 0; tn < 4; ++tn) {
        const float prev = accflag ? cp[tn * 16] : 0.f;
        cp[tn * 16] = acc[half * 4 + tn][r] + bb[tn] + prev;
      }
    }
  }
}

// ---------------------------------------------------------------------------
// GATv2 edge kernels (self-loops appended implicitly: e >= E -> node e-E)
// ---------------------------------------------------------------------------
__global__ void gat_score_kernel(
    const float* __restrict__ xl, const float* __restrict__ xr,
    const long long* __restrict__ e0, const long long* __restrict__ e1,
    const float* __restrict__ att, float* __restrict__ score, int G, int E,
    int Etot, int nNodes, int heads, int D) {
  const int lane = threadIdx.x & 31;
  const long long gw =
      (long long)blockIdx.x * (blockDim.x >> 5) + (threadIdx.x >> 5);
  const long long total = (long long)G * Etot * heads;
  if (gw >= total) return;
  const int hh = (int)(gw % heads);
  const long long ge = gw / heads;
  const int e = (int)(ge % Etot);
  const int g = (int)(ge / Etot);
  int sN, dN;
  if (e < E) { sN = (int)e0[e]; dN = (int)e1[e]; } else { sN = e - E; dN = sN; }
  const int F = heads * D;
  const float* pl = xl + ((size_t)g * nNodes + sN) * F + hh * D;
  const float* pr = xr + ((size_t)g * nNodes + dN) * F + hh * D;
  const float* pa = att + (size_t)hh * D;
  float acc = 0.f;
  for (int d = lane; d < D; d += 32) {
    float v = pl[d] + pr[d];
    v = v > 0.f ? v : 0.2f * v;  // leaky relu
    acc += pa[d] * v;
  }
  for (int o = 16; o > 0; o >>= 1) acc += __shfl_xor(acc, o, 32);
  if (lane == 0) score[gw] = acc;
}

__global__ void gat_max_kernel(const float* __restrict__ score,
                               const long long* __restrict__ e1,
                               float* __restrict__ nmax, int G, int E, int Etot,
                               int nNodes, int heads) {
  long long i = (long long)blockIdx.x * blockDim.x + threadIdx.x;
  const long long total = (long long)G * Etot * heads;
  if (i >= total) return;
  const int hh = (int)(i % heads);
  const long long ge = i / heads;
  const int e = (int)(ge % Etot);
  const int g = (int)(ge / Etot);
  const int dN = (e < E) ? (int)e1[e] : (e - E);
  atomicMaxF(&nmax[((size_t)g * nNodes + dN) * heads + hh], score[i]);
}

__global__ void gat_exp_kernel(float* __restrict__ score,
                               const long long* __restrict__ e1,
                               const float* __restrict__ nmax,
                               float* __restrict__ nden, int G, int E, int Etot,
                               int nNodes, int heads) {
  long long i = (long long)blockIdx.x * blockDim.x + threadIdx.x;
  const long long total = (long long)G * Etot * heads;
  if (i >= total) return;
  const int hh = (int)(i % heads);
  const long long ge = i / heads;
  const int e = (int)(ge % Etot);
  const int g = (int)(ge / Etot);
  const int dN = (e < E) ? (int)e1[e] : (e - E);
  const size_t ni = ((size_t)g * nNodes + dN) * heads + hh;
  const float ex = expf(score[i] - nmax[ni]);
  score[i] = ex;
  atomicAdd(&nden[ni], ex);
}

__global__ void gat_aggr_kernel(
    const float* __restrict__ xl, const float* __restrict__ score,
    const float* __restrict__ nden, const long long* __restrict__ e0,
    const long long* __restrict__ e1, float* __restrict__ outp, int G, int E,
    int Etot, int nNodes, int heads, int D) {
  const int lane = threadIdx.x & 31;
  const long long gw =
      (long long)blockIdx.x * (blockDim.x >> 5) + (threadIdx.x >> 5);
  const long long total = (long long)G * Etot * heads;
  if (gw >= total) return;
  const int hh = (int)(gw % heads);
  const long long ge = gw / heads;
  const int e = (int)(ge % Etot);
  const int g = (int)(ge / Etot);
  int sN, dN;
  if (e < E) { sN = (int)e0[e]; dN = (int)e1[e]; } else { sN = e - E; dN = sN; }
  const int F = heads * D;
  const float alpha =
      score[gw] / (nden[((size_t)g * nNodes + dN) * heads + hh] + 1e-16f);
  const float* pl = xl + ((size_t)g * nNodes + sN) * F + hh * D;
  float* po = outp + ((size_t)g * nNodes + dN) * F + hh * D;
  for (int d = lane; d < D; d += 32) atomicAdd(&po[d], alpha * pl[d]);
}

// h = elu( bn( h + conv_bias ) )  over [rows, F]
__global__ void bias_bn_elu_kernel(float* __restrict__ h,
                                   const float* __restrict__ bconv,
                                   const float* __restrict__ gg,
                                   const float* __restrict__ be,
                                   const float* __restrict__ mm,
                                   const float* __restrict__ vv,
                                   long long total, int F) {
  long long i = (long long)blockIdx.x * blockDim.x + threadIdx.x;
  if (i >= total) return;
  const int c = (int)(i % F);
  float v = h[i] + bconv[c];
  v = (v - mm[c]) * rsqrtf(vv[c] + 1e-5f) * gg[c] + be[c];
  h[i] = v > 0.f ? v : (expf(v) - 1.f);
}

// ---------------------------------------------------------------------------
// Fused LSTM step: one wave owns a 16-row x 16-hidden-col tile; computes all 4
// gate tiles via WMMA (x@Wih^T + h@Whh^T), then the cell update in-register.
// Software-pipelined like the GEMM. REQUIRES M % 16 == 0, H % 4 == 0.
// ---------------------------------------------------------------------------
__global__ __launch_bounds__(128) void lstm_step_kernel(
    const float* __restrict__ Abase, int lda, int gatherMode, int nNodes, int T,
    int ti, const float* __restrict__ Wih, const float* __restrict__ Whh,
    const float* __restrict__ bih, const float* __restrict__ bhh,
    const float* __restrict__ hprev, float* __restrict__ hnext,
    float* __restrict__ cstate, float* __restrict__ outbuf, int M, int H) {
  const int wave = threadIdx.x >> 5;
  const int lane = threadIdx.x & 31;
  const int m0 = (blockIdx.x * 4 + wave) * 16;
  const int cb = blockIdx.y * 16;
  if (m0 >= M) return;
  const int hi = lane >> 4;
  const int l15 = lane & 15;
  const int s = m0 + l15;  // A-operand row for this lane
  int arow;
  if (gatherMode) {
    const int bi2 = s / nNodes;
    const int ni = s - bi2 * nNodes;
    arow = (bi2 * T + ti) * nNodes + ni;  // faithful .view reinterpretation
  } else {
    arow = s;
  }
  const float* ap = Abase + (size_t)arow * lda + hi * 2;
  const float* hp = hprev + (size_t)s * H + hi * 2;
  const int col = cb + l15;  // B-operand column (hidden unit within gate)
  const float* wp[8];
  wp[0] = Wih + (size_t)col * H + hi * 2;
  wp[1] = Wih + (size_t)(col + H) * H + hi * 2;
  wp[2] = Wih + (size_t)(col + 2 * H) * H + hi * 2;
  wp[3] = Wih + (size_t)(col + 3 * H) * H + hi * 2;
  wp[4] = Whh + (size_t)col * H + hi * 2;
  wp[5] = Whh + (size_t)(col + H) * H + hi * 2;
  wp[6] = Whh + (size_t)(col + 2 * H) * H + hi * 2;
  wp[7] = Whh + (size_t)(col + 3 * H) * H + hi * 2;

  v8f acc[4];
#pragma unroll
  for (int i = 0; i < 4; ++i) acc[i] = zero8();

  // pipeline registers
  v2f ac = ld2(ap);
  v2f ahc = ld2(hp);
  v2f wc[8];
#pragma unroll
  for (int i = 0; i < 8; ++i) wc[i] = ld2(wp[i]);

  for (int k = 0; k < H - 4; k += 4) {
    const int kn = k + 4;
    const v2f an = ld2(ap + kn);
    const v2f ahn = ld2(hp + kn);
    v2f wn[8];
#pragma unroll
    for (int i = 0; i < 8; ++i) wn[i] = ld2(wp[i] + kn);
#pragma unroll
    for (int i = 0; i < 4; ++i) acc[i] = wmma4(ac, wc[i], acc[i]);
#pragma unroll
    for (int i = 0; i < 4; ++i) acc[i] = wmma4(ahc, wc[4 + i], acc[i]);
    ac = an;
    ahc = ahn;
#pragma unroll
    for (int i = 0; i < 8; ++i) wc[i] = wn[i];
  }
#pragma unroll
  for (int i = 0; i < 4; ++i) acc[i] = wmma4(ac, wc[i], acc[i]);
#pragma unroll
  for (int i = 0; i < 4; ++i) acc[i] = wmma4(ahc, wc[4 + i], acc[i]);

  const float bI = bih[col] + bhh[col];
  const float bF = bih[col + H] + bhh[col + H];
  const float bG = bih[col + 2 * H] + bhh[col + 2 * H];
  const float bO = bih[col + 3 * H] + bhh[col + 3 * H];
#pragma unroll
  for (int r = 0; r < 8; ++r) {
    const int row = m0 + r + hi * 8;
    const size_t ci = (size_t)row * H + col;
    const float iv = sigmoidf(acc[0][r] + bI);
    const float fv = sigmoidf(acc[1][r] + bF);
    const float gv = tanhf(acc[2][r] + bG);
    const float ov = sigmoidf(acc[3][r] + bO);
    const float cn = fv * cstate[ci] + iv * gv;
    cstate[ci] = cn;
    const float hv = ov * tanhf(cn);
    hnext[ci] = hv;
    outbuf[((size_t)row * T + ti) * H + col] = hv;
  }
}

// ---------------------------------------------------------------------------
// temporal attention pooling: one wave per sequence
// ---------------------------------------------------------------------------
__global__ void attn_pool_kernel(const float* __restrict__ out1,
                                 const float* __restrict__ aW1,
                                 const float* __restrict__ ab1,
                                 const float* __restrict__ aW2,
                                 const float* __restrict__ ab2,
                                 float* __restrict__ ctx, int M, int T, int H,
                                 int Hh) {
  const int lane = threadIdx.x & 31;
  const int s = blockIdx.x * (blockDim.x >> 5) + (threadIdx.x >> 5);
  if (s >= M) return;
  const float* base = out1 + (size_t)s * T * H;
  float sc[16];
  for (int ti = 0; ti < T; ++ti) {
    const float* row = base + (size_t)ti * H;
    float part = 0.f;
    for (int j = lane; j < Hh; j += 32) {
      float a = ab1[j];
      for (int k = 0; k < H; ++k) a += row[k] * aW1[k * Hh + j];
      part += tanhf(a) * aW2[j];
    }
    for (int o = 16; o > 0; o >>= 1) part += __shfl_xor(part, o, 32);
    sc[ti] = part + ab2[0];
  }
  float mx = sc[0];
  for (int ti = 1; ti < T; ++ti) mx = fmaxf(mx, sc[ti]);
  float den = 0.f;
  for (int ti = 0; ti < T; ++ti) { sc[ti] = expf(sc[ti] - mx); den += sc[ti]; }
  const float inv = 1.f / den;
  for (int d = lane; d < H; d += 32) {
    float a = 0.f;
    for (int ti = 0; ti < T; ++ti) a += sc[ti] * base[(size_t)ti * H + d];
    ctx[(size_t)s * H + d] = a * inv;
  }
}

// ---------------------------------------------------------------------------
// skip MLP + final MLP head: one wave per sequence row
// ---------------------------------------------------------------------------
__global__ void head_kernel(const float* __restrict__ x,
                            const float* __restrict__ ctx,
                            const float* __restrict__ sW1,
                            const float* __restrict__ sb1,
                            const float* __restrict__ sW2,
                            const float* __restrict__ sb2,
                            const float* __restrict__ fW1,
                            const float* __restrict__ fb1,
                            const float* __restrict__ fW2,
                            const float* __restrict__ fb2,
                            float* __restrict__ outp, int T, int nNodes,
                            int fdim, int NBF, int M, int H) {
  const int lane = threadIdx.x & 31;
  const int s = blockIdx.x * (blockDim.x >> 5) + (threadIdx.x >> 5);
  if (s >= M) return;
  const int bi = s / nNodes, ni = s - bi * nNodes;
  const float* xin = x + (((size_t)bi * T + (T - 1)) * nNodes + ni) * fdim;
  // hidden1 = gelu(skip_in @ sW1 + sb1)  (64 units, 2 per lane)
  float hid[2];
#pragma unroll
  for (int jj = 0; jj < 2; ++jj) {
    const int j = lane + jj * 32;
    float a = sb1[j];
    for (int k = 0; k < NBF; ++k) a += xin[k] * sW1[k * 64 + j];
    hid[jj] = geluf(a);
  }
  // skip = hidden1 @ sW2 + sb2  (32 units, 1 per lane)
  float sk = sb2[lane];
#pragma unroll
  for (int k = 0; k < 64; ++k) {
    const float hv = __shfl(hid[k >> 5], k & 31, 32);
    sk += hv * sW2[k * 32 + lane];
  }
  // hidden2 = gelu([ctx, skip] @ fW1 + fb1)  (64 units)
  const float* cp = ctx + (size_t)s * H;
  float h2[2];
#pragma unroll
  for (int jj = 0; jj < 2; ++jj) {
    const int j = lane + jj * 32;
    float a = fb1[j];
    for (int k = 0; k < H; ++k) a += cp[k] * fW1[k * 64 + j];
#pragma unroll
    for (int k = 0; k < 32; ++k)
      a += __shfl(sk, k, 32) * fW1[(H + k) * 64 + j];
    h2[jj] = geluf(a);
  }
  float part = h2[0] * fW2[lane] + h2[1] * fW2[lane + 32];
  for (int o = 16; o > 0; o >>= 1) part += __shfl_xor(part, o, 32);
  if (lane == 0) outp[s] = part + fb2[0];
}

// ---------------------------------------------------------------------------
static inline int ceildiv(long long a, long long b) { return (int)((a + b - 1) / b); }

extern "C" void kernel_launch(void* const* d_in, const int* in_sizes, int n_in,
                              void* d_out, int out_size, void* d_ws,
                              size_t ws_size, hipStream_t stream) {
  (void)in_sizes; (void)n_in; (void)out_size; (void)ws_size;
  const float* x = (const float*)d_in[0];
  const long long* ei = (const long long*)d_in[1];
  const float* Wl1 = (const float*)d_in[2];
  const float* Wr1 = (const float*)d_in[3];
  const float* att1 = (const float*)d_in[4];
  const float* b1 = (const float*)d_in[5];
  const float* bn1_g = (const float*)d_in[6];
  const float* bn1_b = (const float*)d_in[7];
  const float* bn1_m = (const float*)d_in[8];
  const float* bn1_v = (const float*)d_in[9];
  const float* Wl2 = (const float*)d_in[10];
  const float* Wr2 = (const float*)d_in[11];
  const float* att2 = (const float*)d_in[12];
  const float* b2 = (const float*)d_in[13];
  const float* bn2_g = (const float*)d_in[14];
  const float* bn2_b = (const float*)d_in[15];
  const float* bn2_m = (const float*)d_in[16];
  const float* bn2_v = (const float*)d_in[17];
  const float* proj_W = (const float*)d_in[18];
  const float* proj_b = (const float*)d_in[19];
  const float* Wih0 = (const float*)d_in[20];
  const float* Whh0 = (const float*)d_in[21];
  const float* bih0 = (const float*)d_in[22];
  const float* bhh0 = (const float*)d_in[23];
  const float* Wih1 = (const float*)d_in[24];
  const float* Whh1 = (const float*)d_in[25];
  const float* bih1 = (const float*)d_in[26];
  const float* bhh1 = (const float*)d_in[27];
  const float* aW1 = (const float*)d_in[28];
  const float* ab1 = (const float*)d_in[29];
  const float* aW2 = (const float*)d_in[30];
  const float* ab2 = (const float*)d_in[31];
  const float* sW1 = (const float*)d_in[32];
  const float* sb1 = (const float*)d_in[33];
  const float* sW2 = (const float*)d_in[34];
  const float* sb2 = (const float*)d_in[35];
  const float* fW1 = (const float*)d_in[36];
  const float* fb1 = (const float*)d_in[37];
  const float* fW2 = (const float*)d_in[38];
  const float* fb2 = (const float*)d_in[39];

  const int b = 2, t = 16, n = 2000, f = 32, H = 128, heads = 2;
  const int E = 16000, Etot = E + n;  // self-loops appended
  const int G = b * t;                // 32 graphs
  const int M1 = G * n;               // 64000 node-rows (mult of 32)
  const int SEQ = b * n;              // 4000 sequences
  const long long* e0 = ei;
  const long long* e1 = ei + E;

  float* ws = (float*)d_ws;
  float* buf0 = ws;                 // 16,384,000
  float* buf1 = buf0 + 16384000;    // 16,384,000
  float* buf2 = buf1 + 16384000;    // 16,384,000
  float* scoreB = buf2 + 16384000;  // 1,152,000
  float* nmax = scoreB + 1152000;   // 128,000
  float* nden = nmax + 128000;      // 128,000
  float* hA = nden + 128000;        // 512,000
  float* hB = hA + 512000;          // 512,000
  float* cS = hB + 512000;          // 512,000
  float* ctx = cS + 512000;         // 512,000

  // ---------------- GAT layer 1 ----------------
  gemm_wmma_kernel<<<dim3(ceildiv(M1, 128), (heads * H) / 64), 128, 0,
                     stream>>>(x, f, Wl1, heads * H, buf0, heads * H, M1,
                               heads * H, f, nullptr, 0);
  gemm_wmma_kernel<<<dim3(ceildiv(M1, 128), (heads * H) / 64), 128, 0,
                     stream>>>(x, f, Wr1, heads * H, buf1, heads * H, M1,
                               heads * H, f, nullptr, 0);

  const long long ag1 = (long long)M1 * heads * H;  // 16,384,000
  fill_kernel<<<ceildiv(ag1, 256), 256, 0, stream>>>(buf2, ag1, 0.f);
  const long long nm1 = (long long)G * n * heads;  // 128,000
  fill_kernel<<<ceildiv(nm1, 256), 256, 0, stream>>>(nmax, nm1, -3.0e38f);
  fill_kernel<<<ceildiv(nm1, 256), 256, 0, stream>>>(nden, nm1, 0.f);

  const long long tw1 = (long long)G * Etot * heads;  // 1,152,000 edge-heads
  gat_score_kernel<<<ceildiv(tw1, 4), 128, 0, stream>>>(
      buf0, buf1, e0, e1, att1, scoreB, G, E, Etot, n, heads, H);
  gat_max_kernel<<<ceildiv(tw1, 256), 256, 0, stream>>>(scoreB, e1, nmax, G, E,
                                                        Etot, n, heads);
  gat_exp_kernel<<<ceildiv(tw1, 256), 256, 0, stream>>>(scoreB, e1, nmax, nden,
                                                        G, E, Etot, n, heads);
  gat_aggr_kernel<<<ceildiv(tw1, 4), 128, 0, stream>>>(
      buf0, scoreB, nden, e0, e1, buf2, G, E, Etot, n, heads, H);
  bias_bn_elu_kernel<<<ceildiv(ag1, 256), 256, 0, stream>>>(
      buf2, b1, bn1_g, bn1_b, bn1_m, bn1_v, ag1, heads * H);

  // ---------------- GAT layer 2 ----------------
  gemm_wmma_kernel<<<dim3(ceildiv(M1, 128), H / 64), 128, 0, stream>>>(
      buf2, heads * H, Wl2, H, buf0, H, M1, H, heads * H, nullptr, 0);
  gemm_wmma_kernel<<<dim3(ceildiv(M1, 128), H / 64), 128, 0, stream>>>(
      buf2, heads * H, Wr2, H, buf1, H, M1, H, heads * H, nullptr, 0);

  const long long ag2 = (long long)M1 * H;  // 8,192,000
  fill_kernel<<<ceildiv(ag2, 256), 256, 0, stream>>>(buf2, ag2, 0.f);
  const long long nm2 = (long long)G * n;  // 64,000
  fill_kernel<<<ceildiv(nm2, 256), 256, 0, stream>>>(nmax, nm2, -3.0e38f);
  fill_kernel<<<ceildiv(nm2, 256), 256, 0, stream>>>(nden, nm2, 0.f);

  const long long tw2 = (long long)G * Etot;  // 576,000
  gat_score_kernel<<<ceildiv(tw2, 4), 128, 0, stream>>>(
      buf0, buf1, e0, e1, att2, scoreB, G, E, Etot, n, 1, H);
  gat_max_kernel<<<ceildiv(tw2, 256), 256, 0, stream>>>(scoreB, e1, nmax, G, E,
                                                        Etot, n, 1);
  gat_exp_kernel<<<ceildiv(tw2, 256), 256, 0, stream>>>(scoreB, e1, nmax, nden,
                                                        G, E, Etot, n, 1);
  gat_aggr_kernel<<<ceildiv(tw2, 4), 128, 0, stream>>>(
      buf0, scoreB, nden, e0, e1, buf2, G, E, Etot, n, 1, H);
  bias_bn_elu_kernel<<<ceildiv(ag2, 256), 256, 0, stream>>>(
      buf2, b2, bn2_g, bn2_b, bn2_m, bn2_v, ag2, H);
  // residual projection: h += x @ proj_W + proj_b
  gemm_wmma_kernel<<<dim3(ceildiv(M1, 128), H / 64), 128, 0, stream>>>(
      x, f, proj_W, H, buf2, H, M1, H, f, proj_b, 1);

  // ---------------- LSTM layer 0 (gathered .view input) ----------------
  const long long st = (long long)SEQ * H;  // 512,000
  fill_kernel<<<ceildiv(st, 256), 256, 0, stream>>>(hA, st, 0.f);
  fill_kernel<<<ceildiv(st, 256), 256, 0, stream>>>(cS, st, 0.f);
  for (int ti = 0; ti < t; ++ti) {
    float* hp = (ti & 1) ? hB : hA;
    float* hn = (ti & 1) ? hA : hB;
    lstm_step_kernel<<<dim3(ceildiv(SEQ, 64), H / 16), 128, 0, stream>>>(
        buf2, H, 1, n, t, ti, Wih0, Whh0, bih0, bhh0, hp, hn, cS, buf0, SEQ, H);
  }
  // ---------------- LSTM layer 1 ----------------
  fill_kernel<<<ceildiv(st, 256), 256, 0, stream>>>(hA, st, 0.f);
  fill_kernel<<<ceildiv(st, 256), 256, 0, stream>>>(cS, st, 0.f);
  for (int ti = 0; ti < t; ++ti) {
    float* hp = (ti & 1) ? hB : hA;
    float* hn = (ti & 1) ? hA : hB;
    lstm_step_kernel<<<dim3(ceildiv(SEQ, 64), H / 16), 128, 0, stream>>>(
        buf0 + (size_t)ti * H, t * H, 0, n, t, ti, Wih1, Whh1, bih1, bhh1, hp,
        hn, cS, buf1, SEQ, H);
  }

  // ---------------- attention pooling + heads ----------------
  attn_pool_kernel<<<ceildiv(SEQ, 4), 128, 0, stream>>>(
      buf1, aW1, ab1, aW2, ab2, ctx, SEQ, t, H, 64);
  head_kernel<<<ceildiv(SEQ, 4), 128, 0, stream>>>(
      x, ctx, sW1, sb1, sW2, sb2, fW1, fb1, fW2, fb2, (float*)d_out, t, n, f,
      13, SEQ, H);
}